// NTXentLoss_61349312856160
// MI455X (gfx1250) — compile-verified
//
#include <hip/hip_runtime.h>

// NT-Xent loss for MI455X (gfx1250, wave32).
// Gram matrix via V_WMMA_F32_16X16X4_F32 (fp32 matches reference precision;
// 17 GFLOP total, 4 MB working set resident in L2 -> compute/L2-bound).
// Logits bounded by 1/T = 2  =>  one-pass exp-sum logsumexp (no running max).
// Per wave: two 16-row M-tiles share one B stream (2 independent WMMA chains,
// 16 FLOP per L2 byte), B fragments explicitly double-buffered in groups of 8
// so >= 8 global_load_b64 are in flight under the WMMAs.

typedef float v2f __attribute__((ext_vector_type(2)));
typedef float v8f __attribute__((ext_vector_type(8)));

#define D        128
#define TEMP_INV 2.0f      // 1 / 0.5
#define ROW_PAD  132       // LDS row pitch: 132 % 64 = 4 banks -> conflict-free
#define BAND     32        // rows per workgroup (two 16-row M-tiles per wave)

// ---------------------------------------------------------------------------
// 1) Row-wise L2 normalize: one wave per row (32 lanes x float4 = 128 floats)
// ---------------------------------------------------------------------------
__global__ __launch_bounds__(256)
void ntxent_normalize(const float* __restrict__ zi,
                      const float* __restrict__ zj,
                      float* __restrict__ zn, int B) {
  const int wave = threadIdx.x >> 5;
  const int lane = threadIdx.x & 31;
  const int row  = blockIdx.x * 8 + wave;
  const int n    = 2 * B;
  if (row >= n) return;

  const float* src = (row < B) ? (zi + (size_t)row * D)
                               : (zj + (size_t)(row - B) * D);
  float4 v = *(const float4*)(src + lane * 4);
  float ss = v.x * v.x + v.y * v.y + v.z * v.z + v.w * v.w;
  #pragma unroll
  for (int off = 16; off > 0; off >>= 1) ss += __shfl_xor(ss, off, 32);

  const float nrm   = sqrtf(ss);
  const float scale = 1.0f / fmaxf(nrm, 1e-12f);
  float4 o;
  o.x = v.x * scale; o.y = v.y * scale; o.z = v.z * scale; o.w = v.w * scale;
  *(float4*)(zn + (size_t)row * D + lane * 4) = o;
}

// ---------------------------------------------------------------------------
// 2) Gram tiles + per-row exp-sum.
// ---------------------------------------------------------------------------
__global__ __launch_bounds__(512)
void ntxent_gram(const float* __restrict__ zn,
                 float* __restrict__ rowsum, int n) {
  __shared__ float As[BAND * ROW_PAD];   // 32 x 128 A band, padded pitch
  __shared__ float red[16][BAND];        // [wave][local row] partial sums

  const int tid    = threadIdx.x;
  // wave index is wave-uniform: pin it to an SGPR so the tj loop is scalar
  // (s_cbranch, no per-iteration EXEC masking).
  const int wave   = __builtin_amdgcn_readfirstlane(tid >> 5);
  const int lane   = tid & 31;
  const int i_base = blockIdx.x * BAND;

  // Stage A band: 4096 floats, 512 threads x 2 float4 (16B-aligned in LDS).
  #pragma unroll
  for (int t = 0; t < 2; ++t) {
    const int idx = (tid + t * 512) * 4;
    const int r = idx >> 7, c = idx & 127;
    float4 v = *(const float4*)(zn + (size_t)(i_base + r) * D + c);
    *(float4*)(&As[r * ROW_PAD + c]) = v;
  }
  __syncthreads();

  // fp32 16x16x4 fragment addressing (A and B share the same per-lane form):
  //   lanes 0-15 : row = lane,    K = k0+0, k0+1
  //   lanes 16-31: row = lane-16, K = k0+2, k0+3
  const int half = lane >> 4;
  const int hl   = lane & 15;
  const float* arow0 = &As[hl * ROW_PAD + 2 * half];          // M-tile 0
  const float* arow1 = &As[(16 + hl) * ROW_PAD + 2 * half];   // M-tile 1

  float acc0[8], acc1[8];
  #pragma unroll
  for (int q = 0; q < 8; ++q) { acc0[q] = 0.0f; acc1[q] = 0.0f; }

  const int ntile = n >> 4;
  for (int tj = wave; tj < ntile; tj += 16) {
    const int j_base = tj << 4;
    const float* brow = zn + (size_t)(j_base + hl) * D + 2 * half;

    // Speculative prefetch of this wave's next column tile (stride 256 rows).
    // No guard needed: speculative prefetches past the live region are
    // silently dropped by hardware.
    __builtin_prefetch(zn + (size_t)(j_base + 256 + hl) * D, 0, 3);

    v8f c0 = {0.f, 0.f, 0.f, 0.f, 0.f, 0.f, 0.f, 0.f};
    v8f c1 = {0.f, 0.f, 0.f, 0.f, 0.f, 0.f, 0.f, 0.f};

    // B stream double-buffered in groups of 8 k-steps: group g+1's loads are
    // issued before group g's WMMAs, keeping 8 loads in flight.
    v2f bb[8];
    #pragma unroll
    for (int t = 0; t < 8; ++t) bb[t] = *(const v2f*)(brow + 4 * t);

    #pragma unroll
    for (int g = 0; g < 4; ++g) {
      v2f bn[8];
      if (g < 3) {
        #pragma unroll
        for (int t = 0; t < 8; ++t)
          bn[t] = *(const v2f*)(brow + (g + 1) * 32 + 4 * t);
      }
      #pragma unroll
      for (int t = 0; t < 8; ++t) {
        const int k0 = g * 32 + 4 * t;
        v2f a0 = *(const v2f*)(arow0 + k0);
        v2f a1 = *(const v2f*)(arow1 + k0);
        c0 = __builtin_amdgcn_wmma_f32_16x16x4_f32(false, a0, false, bb[t],
                                                   (short)0, c0, false, false);
        c1 = __builtin_amdgcn_wmma_f32_16x16x4_f32(false, a1, false, bb[t],
                                                   (short)0, c1, false, false);
      }
      if (g < 3) {
        #pragma unroll
        for (int t = 0; t < 8; ++t) bb[t] = bn[t];   // register rename
      }
    }

    // C layout: lanes 0-15 -> M = q, N = lane; lanes 16-31 -> M = q+8, N = lane-16
    const int col = j_base + hl;
    #pragma unroll
    for (int q = 0; q < 8; ++q) {
      const int r0 = i_base + q + 8 * half;
      const int r1 = r0 + 16;
      const float e0 = __expf(TEMP_INV * c0[q]);
      const float e1 = __expf(TEMP_INV * c1[q]);
      acc0[q] += (r0 == col) ? 0.0f : e0;   // diagonal -> exp(-inf) = 0
      acc1[q] += (r1 == col) ? 0.0f : e1;
    }
  }

  // Reduce across each 16-lane half (xor 1,2,4,8 stays within the half).
  #pragma unroll
  for (int q = 0; q < 8; ++q) {
    float s0 = acc0[q], s1 = acc1[q];
    s0 += __shfl_xor(s0, 1, 32); s1 += __shfl_xor(s1, 1, 32);
    s0 += __shfl_xor(s0, 2, 32); s1 += __shfl_xor(s1, 2, 32);
    s0 += __shfl_xor(s0, 4, 32); s1 += __shfl_xor(s1, 4, 32);
    s0 += __shfl_xor(s0, 8, 32); s1 += __shfl_xor(s1, 8, 32);
    acc0[q] = s0; acc1[q] = s1;
  }
  if (hl == 0) {
    #pragma unroll
    for (int q = 0; q < 8; ++q) {
      red[wave][q + 8 * half]      = acc0[q];
      red[wave][16 + q + 8 * half] = acc1[q];
    }
  }
  __syncthreads();

  // Deterministic fixed-order combine across the 16 waves; exclusive row
  // ownership per workgroup -> plain global store, no atomics.
  if (tid < BAND) {
    float s = 0.0f;
    #pragma unroll
    for (int w = 0; w < 16; ++w) s += red[w][tid];
    rowsum[i_base + tid] = s;
  }
}

// ---------------------------------------------------------------------------
// 3) loss = mean_i( log(rowsum_i) - pos_i ), pos_i = 2 * <zn_i, zn_pair(i)>
// ---------------------------------------------------------------------------
__global__ __launch_bounds__(1024)
void ntxent_final(const float* __restrict__ zn,
                  const float* __restrict__ rowsum,
                  float* __restrict__ out, int B) {
  __shared__ float wred[32];
  const int tid  = threadIdx.x;
  const int wave = tid >> 5;
  const int lane = tid & 31;
  const int n    = 2 * B;

  float local = 0.0f;
  for (int r = wave; r < n; r += 32) {
    const int pr = (r < B) ? (r + B) : (r - B);
    const float4 a = *(const float4*)(zn + (size_t)r  * D + lane * 4);
    const float4 b = *(const float4*)(zn + (size_t)pr * D + lane * 4);
    float d = a.x * b.x + a.y * b.y + a.z * b.z + a.w * b.w;
    #pragma unroll
    for (int off = 16; off > 0; off >>= 1) d += __shfl_xor(d, off, 32);
    const float pos = TEMP_INV * d;
    const float lse = __logf(rowsum[r]);
    local += (lse - pos);
  }
  if (lane == 0) wred[wave] = local;
  __syncthreads();
  if (tid == 0) {
    float s = 0.0f;
    #pragma unroll
    for (int w = 0; w < 32; ++w) s += wred[w];
    out[0] = s / (float)n;
  }
}

// ---------------------------------------------------------------------------
extern "C" void kernel_launch(void* const* d_in, const int* in_sizes, int n_in,
                              void* d_out, int out_size, void* d_ws, size_t ws_size,
                              hipStream_t stream) {
  (void)n_in; (void)out_size; (void)ws_size;
  const float* zi = (const float*)d_in[0];
  const float* zj = (const float*)d_in[1];
  const int B = in_sizes[0] / D;   // 4096
  const int n = 2 * B;             // 8192

  float* zn     = (float*)d_ws;                 // n x D fp32   (4 MB)
  float* rowsum = zn + (size_t)n * D;           // n   fp32     (32 KB)

  ntxent_normalize<<<n / 8, 256, 0, stream>>>(zi, zj, zn, B);
  ntxent_gram<<<n / BAND, 512, 0, stream>>>(zn, rowsum, n);
  ntxent_final<<<1, 1024, 0, stream>>>(zn, rowsum, (float*)d_out, B);
}